// EMA_1D_v2_83949430767675
// MI455X (gfx1250) — compile-verified
//
#include <hip/hip_runtime.h>

typedef __attribute__((ext_vector_type(2))) float v2f;
typedef __attribute__((ext_vector_type(8))) float v8f;

#define NGROUPS 512          // b * GROUPS = 16 * 32
#define CG 16                // channels per group
#define LLEN 8192            // sequence length
#define POS_PER_WAVE 256
#define TILES_PER_WAVE (POS_PER_WAVE / 16)       // 16
#define WAVES_PER_GROUP (LLEN / POS_PER_WAVE)    // 32

// ---------------------------------------------------------------------------
// Kernel 1: per-group channel sums  S[g,c] = sum_t x[g,c,t]  (deterministic)
// float4 loads: 8 sweeps x 16 channels x b128 per thread.
// ---------------------------------------------------------------------------
__global__ __launch_bounds__(256) void k_chansum(const float* __restrict__ x,
                                                 float* __restrict__ S) {
  int g = blockIdx.x;
  const float* xg = x + (size_t)g * CG * LLEN;
  float acc[CG];
#pragma unroll
  for (int c = 0; c < CG; ++c) acc[c] = 0.f;
  for (int t = threadIdx.x * 4; t < LLEN; t += 256 * 4) {
#pragma unroll
    for (int c = 0; c < CG; ++c) {
      const float4 v = *(const float4*)(xg + (size_t)c * LLEN + t);
      acc[c] += (v.x + v.y) + (v.z + v.w);
    }
  }
#pragma unroll
  for (int c = 0; c < CG; ++c) {
#pragma unroll
    for (int d = 16; d >= 1; d >>= 1) acc[c] += __shfl_xor(acc[c], d, 32);
  }
  __shared__ float sm[8][CG];
  int wave = threadIdx.x >> 5;
  int lane = threadIdx.x & 31;
  if (lane == 0) {
#pragma unroll
    for (int c = 0; c < CG; ++c) sm[wave][c] = acc[c];
  }
  __syncthreads();
  if (threadIdx.x < CG) {
    float s = 0.f;
#pragma unroll
    for (int w = 0; w < 8; ++w) s += sm[w][threadIdx.x];
    S[g * CG + threadIdx.x] = s;
  }
}

// ---------------------------------------------------------------------------
// Kernel 2: conv means (exact, via linearity + edge correction) -> softmax
//           a1[g,o], a2[g,o], and cb[g] = a1.b3 + a2.b1
// ---------------------------------------------------------------------------
__global__ __launch_bounds__(256) void k_attn(const float* __restrict__ x,
                                              const float* __restrict__ w1,
                                              const float* __restrict__ b1,
                                              const float* __restrict__ w3,
                                              const float* __restrict__ b3,
                                              const float* __restrict__ S,
                                              float* __restrict__ a1o,
                                              float* __restrict__ a2o,
                                              float* __restrict__ cbo) {
  int idx = blockIdx.x * 256 + threadIdx.x;     // 0 .. NGROUPS*CG-1
  int g = idx >> 4;
  int o = idx & 15;
  const float* xg = x + (size_t)g * CG * LLEN;
  const float invL = 1.0f / (float)LLEN;
  float m1 = 0.f, m2 = 0.f;
#pragma unroll
  for (int i = 0; i < CG; ++i) {
    float Si  = S[g * CG + i];
    float w30 = w3[(o * CG + i) * 3 + 0];
    float w31 = w3[(o * CG + i) * 3 + 1];
    float w32 = w3[(o * CG + i) * 3 + 2];
    float xf  = xg[(size_t)i * LLEN];              // x[g,i,0]
    float xl  = xg[(size_t)i * LLEN + (LLEN - 1)]; // x[g,i,L-1]
    m1 += w1[o * CG + i] * Si;
    m2 += w30 * (Si - xl) + w31 * Si + w32 * (Si - xf);
  }
  m1 = m1 * invL + b1[o];
  m2 = m2 * invL + b3[o];
  // softmax over the 16 lanes sharing this group (xor masks <= 8 stay in half)
  float mx1 = m1, mx2 = m2;
#pragma unroll
  for (int d = 8; d >= 1; d >>= 1) {
    mx1 = fmaxf(mx1, __shfl_xor(mx1, d, 32));
    mx2 = fmaxf(mx2, __shfl_xor(mx2, d, 32));
  }
  float e1 = __expf(m1 - mx1);
  float e2 = __expf(m2 - mx2);
  float s1 = e1, s2 = e2;
#pragma unroll
  for (int d = 8; d >= 1; d >>= 1) {
    s1 += __shfl_xor(s1, d, 32);
    s2 += __shfl_xor(s2, d, 32);
  }
  float a1v = e1 / s1;
  float a2v = e2 / s2;
  a1o[idx] = a1v;
  a2o[idx] = a2v;
  float cb = a1v * b3[o] + a2v * b1[o];
#pragma unroll
  for (int d = 8; d >= 1; d >>= 1) cb += __shfl_xor(cb, d, 32);
  if (o == 0) cbo[g] = cb;
}

// ---------------------------------------------------------------------------
// Kernel 3: main streaming pass. One wave = one group x 256 positions.
// Per 16-position tile: x1 via 4 WMMA f32 16x16x4, x2 via 12 (3 taps),
// weights via per-lane dot + shfl_xor(16), out = x * sigmoid(weights).
// ---------------------------------------------------------------------------
__global__ __launch_bounds__(256) void k_main(const float* __restrict__ x,
                                              const float* __restrict__ w1,
                                              const float* __restrict__ w3,
                                              const float* __restrict__ a1ws,
                                              const float* __restrict__ a2ws,
                                              const float* __restrict__ cbws,
                                              float* __restrict__ out) {
  int waveInBlock = threadIdx.x >> 5;
  int lane = threadIdx.x & 31;
  int gw = blockIdx.x * 8 + waveInBlock;
  int g = gw / WAVES_PER_GROUP;
  int chunk = gw % WAVES_PER_GROUP;
  int h = lane >> 4;    // half-wave select (K split / row-half split)
  int n = lane & 15;    // A: row M; B: column N; C/D: column N

  const float* xg = x + (size_t)g * CG * LLEN;
  float* og = out + (size_t)g * CG * LLEN;

  // A fragments (16x4 f32): lane 16h+m holds A[m][k0+2h], A[m][k0+2h+1]
  v2f A1[4];
  v2f A3[3][4];
#pragma unroll
  for (int q = 0; q < 4; ++q) {
    int k = 4 * q + 2 * h;
    A1[q].x = w1[n * CG + k];
    A1[q].y = w1[n * CG + k + 1];
#pragma unroll
    for (int tap = 0; tap < 3; ++tap) {
      A3[tap][q].x = w3[(n * CG + k) * 3 + tap];
      A3[tap][q].y = w3[(n * CG + k + 1) * 3 + tap];
    }
  }
  // attention coefs for the 8 rows (M = j + 8h) this lane owns in C/D
  float ac1[8], ac2[8];
#pragma unroll
  for (int j = 0; j < 8; ++j) {
    ac1[j] = a1ws[g * CG + j + 8 * h];
    ac2[j] = a2ws[g * CG + j + 8 * h];
  }
  float cbg = cbws[g];

  for (int tile = 0; tile < TILES_PER_WAVE; ++tile) {
    int t0 = chunk * POS_PER_WAVE + tile * 16;
    int tc = t0 + n;

    // center B fragments (4x16 f32): lane 16h+n holds B[k0+2h][n], B[k0+2h+1][n]
    v2f Bc[4];
#pragma unroll
    for (int q = 0; q < 4; ++q) {
      int k = 4 * q + 2 * h;
      Bc[q].x = xg[(size_t)k * LLEN + tc];
      Bc[q].y = xg[(size_t)(k + 1) * LLEN + tc];
    }

    v8f c1 = {0.f, 0.f, 0.f, 0.f, 0.f, 0.f, 0.f, 0.f};
#pragma unroll
    for (int q = 0; q < 4; ++q)
      c1 = __builtin_amdgcn_wmma_f32_16x16x4_f32(false, A1[q], false, Bc[q],
                                                 (short)0, c1, false, false);

    v8f c2 = {0.f, 0.f, 0.f, 0.f, 0.f, 0.f, 0.f, 0.f};
    // left-shifted tap (k=0): positions t-1 with zero pad at t=-1
    {
      int tl = tc - 1;
      bool ok = (tl >= 0);
      v2f Bs[4];
#pragma unroll
      for (int q = 0; q < 4; ++q) {
        int k = 4 * q + 2 * h;
        Bs[q].x = ok ? xg[(size_t)k * LLEN + tl] : 0.f;
        Bs[q].y = ok ? xg[(size_t)(k + 1) * LLEN + tl] : 0.f;
      }
#pragma unroll
      for (int q = 0; q < 4; ++q)
        c2 = __builtin_amdgcn_wmma_f32_16x16x4_f32(false, A3[0][q], false, Bs[q],
                                                   (short)0, c2, false, false);
    }
    // center tap (k=1): reuse Bc
#pragma unroll
    for (int q = 0; q < 4; ++q)
      c2 = __builtin_amdgcn_wmma_f32_16x16x4_f32(false, A3[1][q], false, Bc[q],
                                                 (short)0, c2, false, false);
    // right-shifted tap (k=2): positions t+1 with zero pad at t=L
    {
      int tr = tc + 1;
      bool ok = (tr < LLEN);
      v2f Bs[4];
#pragma unroll
      for (int q = 0; q < 4; ++q) {
        int k = 4 * q + 2 * h;
        Bs[q].x = ok ? xg[(size_t)k * LLEN + tr] : 0.f;
        Bs[q].y = ok ? xg[(size_t)(k + 1) * LLEN + tr] : 0.f;
      }
#pragma unroll
      for (int q = 0; q < 4; ++q)
        c2 = __builtin_amdgcn_wmma_f32_16x16x4_f32(false, A3[2][q], false, Bs[q],
                                                   (short)0, c2, false, false);
    }

    // weights[tc] = a1 . x2[:,tc] + a2 . x1[:,tc] + (a1.b3 + a2.b1)
    float p = 0.f;
#pragma unroll
    for (int j = 0; j < 8; ++j) p += ac1[j] * c2[j] + ac2[j] * c1[j];
    p += __shfl_xor(p, 16, 32);   // combine row-halves; same column partner
    // fast sigmoid: v_exp_f32 + v_rcp_f32 (no IEEE divide chain)
    float sg = __builtin_amdgcn_rcpf(1.0f + __expf(-(p + cbg)));

    // out[g,k,tc] = x[g,k,tc] * sig  — reuse center B registers (all 16 rows
    // are covered exactly once across the two half-waves); NT stores: the
    // output is write-once, keep it out of L2.
#pragma unroll
    for (int q = 0; q < 4; ++q) {
      int k = 4 * q + 2 * h;
      __builtin_nontemporal_store(Bc[q].x * sg, &og[(size_t)k * LLEN + tc]);
      __builtin_nontemporal_store(Bc[q].y * sg, &og[(size_t)(k + 1) * LLEN + tc]);
    }
  }
}

// ---------------------------------------------------------------------------
extern "C" void kernel_launch(void* const* d_in, const int* in_sizes, int n_in,
                              void* d_out, int out_size, void* d_ws, size_t ws_size,
                              hipStream_t stream) {
  const float* x  = (const float*)d_in[0];
  const float* w1 = (const float*)d_in[1];
  const float* b1 = (const float*)d_in[2];
  const float* w3 = (const float*)d_in[3];
  const float* b3 = (const float*)d_in[4];
  float* out = (float*)d_out;

  float* S  = (float*)d_ws;            // NGROUPS*CG
  float* a1 = S  + NGROUPS * CG;       // NGROUPS*CG
  float* a2 = a1 + NGROUPS * CG;       // NGROUPS*CG
  float* cb = a2 + NGROUPS * CG;       // NGROUPS

  k_chansum<<<NGROUPS, 256, 0, stream>>>(x, S);
  k_attn<<<(NGROUPS * CG) / 256, 256, 0, stream>>>(x, w1, b1, w3, b3, S, a1, a2, cb);
  k_main<<<(NGROUPS * WAVES_PER_GROUP) / 8, 256, 0, stream>>>(x, w1, w3, a1, a2, cb, out);
}